// WindowAttention3D_62543313764644
// MI455X (gfx1250) — compile-verified
//
#include <hip/hip_runtime.h>
#include <hip/hip_bf16.h>
#include <math.h>

typedef __attribute__((ext_vector_type(2))) float v2f;
typedef __attribute__((ext_vector_type(4))) float v4f;
typedef __attribute__((ext_vector_type(8))) float v8f;
typedef __attribute__((ext_vector_type(4))) int   v4i;

#define SCALE 0.17677669529663687f   // 32^-0.5
#define NWIN  1024
#define NTOK  99
#define NPAD  112
#define HEADS 12
#define HDIM  32
#define DIM   384
#define KVSTR 36                      // K/V LDS row stride (16B-aligned, bank-clean)

#define AS1 __attribute__((address_space(1)))
#define AS3 __attribute__((address_space(3)))

#if defined(__has_builtin)
# if __has_builtin(__builtin_amdgcn_global_load_async_to_lds_b128)
#  define HAVE_ASYNC_LDS 1
# endif
#endif

// Copy 16 bytes global -> LDS. Async (ASYNCcnt-tracked DMA) when available.
static __device__ __forceinline__ void copy16_g2l(const float* g, float* l) {
#ifdef HAVE_ASYNC_LDS
    __builtin_amdgcn_global_load_async_to_lds_b128((AS1 v4i*)g, (AS3 v4i*)l, 0, 0);
#else
    *(v4f*)l = *(const v4f*)g;
#endif
}

static __device__ __forceinline__ void wait_async_copies() {
#ifdef HAVE_ASYNC_LDS
# if __has_builtin(__builtin_amdgcn_s_wait_asynccnt)
    __builtin_amdgcn_s_wait_asynccnt(0);
# else
    asm volatile("s_wait_asynccnt 0" ::: "memory");
# endif
#endif
}

// D = A(16x4) * B(4x16) + C, pure fp32 matrix op
static __device__ __forceinline__ v8f wmma_f32(v2f a, v2f b, v8f c) {
    return __builtin_amdgcn_wmma_f32_16x16x4_f32(false, a, false, b, (short)0, c, false, false);
}

// ---------------------------------------------------------------------------
// Kernel 1: materialize padded relative-position bias  [12][99][99]
// ---------------------------------------------------------------------------
__global__ void bias_build(const float* __restrict__ rpb, const int* __restrict__ rpi,
                           float* __restrict__ bias) {
    int idx = blockIdx.x * 256 + threadIdx.x;
    const int total = HEADS * NTOK * NTOK;
    if (idx >= total) return;
    int h = idx / (NTOK * NTOK);
    int rem = idx % (NTOK * NTOK);
    int n = rem / NTOK, m = rem % NTOK;
    float v = 0.f;
    if (n > 0 && m > 0) v = rpb[rpi[(n - 1) * 98 + (m - 1)] * HEADS + h];
    bias[idx] = v;
}

// ---------------------------------------------------------------------------
// Kernel 2: QKV GEMM  [1024*99,384] @ [384,1152] -> Q,K,V as [1024][12][112][32]
// ---------------------------------------------------------------------------
__global__ void __launch_bounds__(128) qkv_gemm(const float* __restrict__ x,
                                                const float* __restrict__ Wqkv,
                                                float* __restrict__ Q,
                                                float* __restrict__ K,
                                                float* __restrict__ V) {
    __shared__ __align__(16) float Xs[16 * 388];   // row stride 1552B: 16B aligned
    const int b = blockIdx.x;
    const int rt = blockIdx.y;                     // 0..6 row tile
    const int t = threadIdx.x, lane = t & 31, wave = t >> 5;

    // stage X tile (16 rows x 384) via async b128; zero-fill padded rows
    for (int i4 = t; i4 < 16 * 96; i4 += 128) {
        int r = i4 / 96, c = (i4 % 96) * 4;
        int n = rt * 16 + r;
        if (n < NTOK) {
            copy16_g2l(x + ((size_t)b * NTOK + n) * DIM + c, &Xs[r * 388 + c]);
        } else {
            v4f z = {};
            *(v4f*)&Xs[r * 388 + c] = z;
        }
    }
    wait_async_copies();
    __syncthreads();

    const int m = lane & 15, hi = lane >> 4;
    for (int ct = wave; ct < 72; ct += 4) {
        int col = ct * 16 + m;                     // 0..1151
        v8f acc = {};
        for (int c = 0; c < 96; ++c) {             // K = 384, step 4
            int k0 = 4 * c + 2 * hi;
            v2f a; a.x = Xs[m * 388 + k0]; a.y = Xs[m * 388 + k0 + 1];
            v2f bb;
            bb.x = Wqkv[(size_t)k0 * 1152 + col];
            bb.y = Wqkv[(size_t)(k0 + 1) * 1152 + col];
            acc = wmma_f32(a, bb, acc);
        }
        int which = col / DIM;                     // 0=q 1=k 2=v
        int cc = col % DIM;
        int h = cc >> 5, d = cc & 31;
        float* dst = (which == 0) ? Q : ((which == 1) ? K : V);
        float sc = (which == 0) ? SCALE : 1.f;
        size_t base = (((size_t)b * HEADS + h) * NPAD) * HDIM + d;
#pragma unroll
        for (int r = 0; r < 8; ++r) {
            int row = rt * 16 + r + 8 * hi;        // 0..111, rows>=99 are zeros
            dst[base + (size_t)row * HDIM] = acc[r] * sc;
        }
    }
}

// ---------------------------------------------------------------------------
// Kernel 3: fused attention per (window, head).
// ---------------------------------------------------------------------------
__global__ void __launch_bounds__(128) attn_kernel(const float* __restrict__ Q,
                                                   const float* __restrict__ K,
                                                   const float* __restrict__ V,
                                                   const float* __restrict__ bias,
                                                   const float* __restrict__ mask,
                                                   float* __restrict__ O) {
    __shared__ __align__(16) float Ks[NPAD * KVSTR];
    __shared__ __align__(16) float Vs[NPAD * KVSTR];
    __shared__ float Ss[4][16 * 113];              // per-wave P strip, odd stride
    const int b = blockIdx.x, h = blockIdx.y;
    const int t = threadIdx.x, lane = t & 31, wave = t >> 5;
    const size_t kvbase = ((size_t)b * HEADS + h) * NPAD * HDIM;

    // stage K,V (112 x 32) via async b128 into padded-stride LDS
    for (int i4 = t; i4 < (NPAD * HDIM) / 4; i4 += 128) {
        int r = i4 >> 3, c = (i4 & 7) << 2;        // 8 float4 per 32-float row
        copy16_g2l(K + kvbase + (size_t)i4 * 4, &Ks[r * KVSTR + c]);
        copy16_g2l(V + kvbase + (size_t)i4 * 4, &Vs[r * KVSTR + c]);
    }
    wait_async_copies();
    __syncthreads();

    const int m = lane & 15, hi = lane >> 4;
    const int wmask = b & 63;                      // window index inside mask batch
    float* strip = &Ss[wave][0];

    for (int rt = wave; rt < 7; rt += 4) {
        // preload Q A-fragments for this 16-row strip (16 floats/lane)
        float areg[16];
        const float* qrow = &Q[kvbase + (size_t)(rt * 16 + m) * HDIM];
#pragma unroll
        for (int c = 0; c < 8; ++c) {
            areg[2 * c]     = qrow[4 * c + 2 * hi];
            areg[2 * c + 1] = qrow[4 * c + 2 * hi + 1];
        }
        // ---- S = Q K^T, spill to strip with bias+mask ----
        for (int ct = 0; ct < 7; ++ct) {
            v8f acc = {};
#pragma unroll
            for (int c = 0; c < 8; ++c) {          // K-dim 32, step 4
                int k0 = 4 * c + 2 * hi;
                v2f a; a.x = areg[2 * c]; a.y = areg[2 * c + 1];
                int ncol = ct * 16 + m;
                v2f bb; bb.x = Ks[ncol * KVSTR + k0]; bb.y = Ks[ncol * KVSTR + k0 + 1];
                acc = wmma_f32(a, bb, acc);
            }
            int col = ct * 16 + m;                 // 0..111
#pragma unroll
            for (int r = 0; r < 8; ++r) {
                int row = rt * 16 + r + 8 * hi;
                float val;
                if (row < NTOK && col < NTOK) {
                    val = acc[r] + bias[(h * NTOK + row) * NTOK + col];
                    if (row > 0 && col > 0)
                        val += mask[((size_t)wmask * 98 + (row - 1)) * 98 + (col - 1)];
                } else if (row >= NTOK) {
                    val = 0.f;                     // dead rows: keep finite
                } else {
                    val = -3.0e38f;                // padded cols: exp -> 0
                }
                strip[(r + 8 * hi) * 113 + col] = val;
            }
        }
        // ---- softmax over strip rows (lane pair per row, shfl_xor 16) ----
        {
            int r0 = m, c0 = hi * 56;
            float mx = -3.4e38f;
            for (int c = 0; c < 56; ++c) mx = fmaxf(mx, strip[r0 * 113 + c0 + c]);
            mx = fmaxf(mx, __shfl_xor(mx, 16));
            float sum = 0.f;
            for (int c = 0; c < 56; ++c) {
                float e = __expf(strip[r0 * 113 + c0 + c] - mx);
                strip[r0 * 113 + c0 + c] = e;
                sum += e;
            }
            sum += __shfl_xor(sum, 16);
            float inv = 1.0f / sum;
            for (int c = 0; c < 56; ++c) strip[r0 * 113 + c0 + c] *= inv;
        }
        // ---- O = P V ----
#pragma unroll
        for (int nt = 0; nt < 2; ++nt) {
            v8f acc = {};
            for (int c = 0; c < 28; ++c) {         // K-dim 112, step 4
                int k0 = 4 * c + 2 * hi;
                v2f a; a.x = strip[m * 113 + k0]; a.y = strip[m * 113 + k0 + 1];
                int d = nt * 16 + m;
                v2f bb; bb.x = Vs[k0 * KVSTR + d]; bb.y = Vs[(k0 + 1) * KVSTR + d];
                acc = wmma_f32(a, bb, acc);
            }
#pragma unroll
            for (int r = 0; r < 8; ++r) {
                int row = rt * 16 + r + 8 * hi;
                if (row < NTOK)
                    O[((size_t)b * NTOK + row) * DIM + h * HDIM + nt * 16 + m] = acc[r];
            }
        }
    }
}

// ---------------------------------------------------------------------------
// Kernel 4: projection GEMM [101376,384]@[384,384] + bias, split cls token out
// ---------------------------------------------------------------------------
__global__ void __launch_bounds__(128) proj_gemm(const float* __restrict__ O,
                                                 const float* __restrict__ Wp,
                                                 const float* __restrict__ bp,
                                                 float* __restrict__ out) {
    __shared__ __align__(16) float Xs[16 * 388];
    const size_t rowbase = (size_t)blockIdx.x * 16;   // 6336 blocks * 16 = 101376
    const int t = threadIdx.x, lane = t & 31, wave = t >> 5;

    for (int i4 = t; i4 < 16 * 96; i4 += 128) {
        int r = i4 / 96, c = (i4 % 96) * 4;
        copy16_g2l(O + (rowbase + r) * DIM + c, &Xs[r * 388 + c]);
    }
    wait_async_copies();
    __syncthreads();

    const int m = lane & 15, hi = lane >> 4;
    for (int ct = wave; ct < 24; ct += 4) {
        int col = ct * 16 + m;
        v8f acc = {};
        for (int c = 0; c < 96; ++c) {
            int k0 = 4 * c + 2 * hi;
            v2f a; a.x = Xs[m * 388 + k0]; a.y = Xs[m * 388 + k0 + 1];
            v2f bb;
            bb.x = Wp[(size_t)k0 * DIM + col];
            bb.y = Wp[(size_t)(k0 + 1) * DIM + col];
            acc = wmma_f32(a, bb, acc);
        }
        float bcol = bp[col];
#pragma unroll
        for (int r = 0; r < 8; ++r) {
            size_t row = rowbase + r + 8 * hi;
            int b = (int)(row / NTOK), n = (int)(row % NTOK);
            float v = acc[r] + bcol;
            if (n == 0)
                out[(size_t)NWIN * 98 * DIM + (size_t)b * DIM + col] = v;   // cls segment
            else
                out[((size_t)b * 98 + (n - 1)) * DIM + col] = v;            // main segment
        }
    }
}

// ---------------------------------------------------------------------------
extern "C" void kernel_launch(void* const* d_in, const int* in_sizes, int n_in,
                              void* d_out, int out_size, void* d_ws, size_t ws_size,
                              hipStream_t stream) {
    const float* x     = (const float*)d_in[0];
    const float* mask  = (const float*)d_in[1];
    const float* Wqkv  = (const float*)d_in[2];
    const float* rpb   = (const float*)d_in[3];
    const float* Wproj = (const float*)d_in[4];
    const float* bproj = (const float*)d_in[5];
    const int*   rpi   = (const int*)d_in[6];

    const size_t QKV_ELEMS = (size_t)NWIN * HEADS * NPAD * HDIM;  // 44,040,192
    const size_t O_ELEMS   = (size_t)NWIN * NTOK * DIM;           // 38,928,384
    float* ws = (float*)d_ws;
    float* Q    = ws;
    float* K    = ws + QKV_ELEMS;
    float* V    = ws + 2 * QKV_ELEMS;
    float* Obuf = ws + 3 * QKV_ELEMS;
    float* bias = Obuf + O_ELEMS;

    bias_build<<<(HEADS * NTOK * NTOK + 255) / 256, 256, 0, stream>>>(rpb, rpi, bias);
    qkv_gemm<<<dim3(NWIN, 7), 128, 0, stream>>>(x, Wqkv, Q, K, V);
    attn_kernel<<<dim3(NWIN, HEADS), 128, 0, stream>>>(Q, K, V, bias, mask, Obuf);
    proj_gemm<<<(NWIN * NTOK) / 16, 128, 0, stream>>>(Obuf, Wproj, bproj, (float*)d_out);
}